// CenterNetDecoder_51410758533788
// MI455X (gfx1250) — compile-verified
//
#include <hip/hip_runtime.h>
#include <stdint.h>

#define BATCH   16
#define NCLS    80
#define HH      128
#define WW      128
#define HW      (HH*WW)
#define K_TOP   100
#define THRV    0.05f
#define STRIDEF 4.0f
#define IMGW    512.0f
#define IMGH    512.0f
#define NSTRIP  16
#define TROWS   8
#define NROWS   (TROWS + 2)        // tile rows incl. halo
#define ROWCH   (WW / 4)           // float4 chunks per row = 32
#define NBIN    32768
#define CAP     4096
#define SORTN   4096

// ---------------------------------------------------------------------------
// CDNA5 async global->LDS copies (ASYNCcnt-tracked), with portable fallbacks.
// ---------------------------------------------------------------------------
#define AS1 __attribute__((address_space(1)))
#define AS3 __attribute__((address_space(3)))

typedef int v4i __attribute__((ext_vector_type(4)));

#if defined(__has_builtin)
#if __has_builtin(__builtin_amdgcn_global_load_async_to_lds_b128)
#define HAVE_ASYNC_B128 1
#endif
#if __has_builtin(__builtin_amdgcn_global_load_async_to_lds_b32)
#define HAVE_ASYNC_B32 1
#endif
#endif

// Copy 16 contiguous bytes global -> LDS. Both pointers 16B aligned.
__device__ __forceinline__ void stage_f128_to_lds(const float* __restrict__ g,
                                                  float* __restrict__ l) {
#if defined(HAVE_ASYNC_B128)
    __builtin_amdgcn_global_load_async_to_lds_b128(
        (AS1 v4i*)(uintptr_t)(const void*)g,
        (AS3 v4i*)(uintptr_t)(void*)l, /*offset=*/0, /*cpol=*/0);
#elif defined(HAVE_ASYNC_B32)
    AS1 int* gp = (AS1 int*)(uintptr_t)(const void*)g;
    AS3 int* lp = (AS3 int*)(uintptr_t)(void*)l;
    // INST_OFFSET is added to BOTH the global and LDS addresses (ISA 10.7).
    __builtin_amdgcn_global_load_async_to_lds_b32(gp, lp, 0,  0);
    __builtin_amdgcn_global_load_async_to_lds_b32(gp, lp, 4,  0);
    __builtin_amdgcn_global_load_async_to_lds_b32(gp, lp, 8,  0);
    __builtin_amdgcn_global_load_async_to_lds_b32(gp, lp, 12, 0);
#else
    *(float4*)l = *(const float4*)g;
#endif
}

__device__ __forceinline__ void wait_async_lds() {
#if defined(HAVE_ASYNC_B128) || defined(HAVE_ASYNC_B32)
#if __has_builtin(__builtin_amdgcn_s_wait_asynccnt)
    __builtin_amdgcn_s_wait_asynccnt(0);
#else
    asm volatile("s_wait_asynccnt 0x0" ::: "memory");
#endif
#endif
}

__device__ __forceinline__ float sigm(float v) {
    return 1.0f / (1.0f + __expf(-v));
}
__device__ __forceinline__ float max3(float a, float b, float c) {
    return fmaxf(a, fmaxf(b, c));
}

// ---------------------------------------------------------------------------
// Tile routine: async-stage raw tile (contiguous rows) into LDS, vectorized
// sigmoid transform in place, register sliding-window 3x3 max test,
// emit(score, flat_index) per survivor.  Tile: 10 x 128 f32 = 5 KB LDS.
// 256 threads, 4 px/thread.
// ---------------------------------------------------------------------------
template <typename F>
__device__ __forceinline__ void nms_tile(const float* __restrict__ hm,
                                         int b, int c, int strip,
                                         float4* sig4, F emit) {
    float* sig = (float*)sig4;
    const float* base = hm + ((size_t)(b * NCLS + c)) * HW;
    const int y0  = strip * TROWS;
    const int gy0 = y0 - 1;                         // global y of tile row 0
    const int tid = threadIdx.x;

    // Valid tile-row range [lo, hi): at most one pad row at each end.
    const int lo = (gy0 < 0) ? 1 : 0;
    const int hi = (gy0 + NROWS > HH) ? (NROWS - 1) : NROWS;

    // Pad rows = -1 (-1 < any sigmoid >= 0 -> identity for the 3x3 max).
    if (lo == 1)
        for (int q = tid; q < ROWCH; q += 256)
            sig4[q] = make_float4(-1.f, -1.f, -1.f, -1.f);
    if (hi == NROWS - 1)
        for (int q = tid; q < ROWCH; q += 256)
            sig4[(NROWS - 1) * ROWCH + q] = make_float4(-1.f, -1.f, -1.f, -1.f);

    // Async-stage the contiguous valid block (16B granules).
    const int nch = (hi - lo) * ROWCH;
    const float* gsrc = base + (gy0 + lo) * WW;
    float* ldst = sig + lo * WW;
    for (int q = tid; q < nch; q += 256)
        stage_f128_to_lds(gsrc + 4 * q, ldst + 4 * q);
    wait_async_lds();
    __syncthreads();

    // Raw -> sigmoid in place, 4 lanes per DS op. Compare sigmoids so that
    // bit-equal rounded sigmoids tie exactly like the reference hmax==hm.
    for (int qq = tid; qq < nch; qq += 256) {
        const int q = lo * ROWCH + qq;
        float4 w = sig4[q];
        w.x = sigm(w.x); w.y = sigm(w.y); w.z = sigm(w.z); w.w = sigm(w.w);
        sig4[q] = w;
    }
    __syncthreads();

    // 3x3 local-max via register sliding windows: per row one aligned float4
    // plus the two edge scalars.
    const int r  = tid >> 5;          // 0..7 within strip
    const int x0 = (tid & 31) * 4;    // 0..124
    float m0 = -1.f, m1 = -1.f, m2 = -1.f, m3 = -1.f;
    float4 cen;
#pragma unroll
    for (int rr = 0; rr < 3; ++rr) {
        const float* row = sig + (r + rr) * WW;
        float4 mv = *(const float4*)(row + x0);
        float lv = (x0 > 0)       ? row[x0 - 1] : -1.0f;
        float rv = (x0 + 4 < WW)  ? row[x0 + 4] : -1.0f;
        m0 = fmaxf(m0, max3(lv,   mv.x, mv.y));
        m1 = fmaxf(m1, max3(mv.x, mv.y, mv.z));
        m2 = fmaxf(m2, max3(mv.y, mv.z, mv.w));
        m3 = fmaxf(m3, max3(mv.z, mv.w, rv));
        if (rr == 1) cen = mv;
    }
    const int flat0 = c * HW + (y0 + r) * WW + x0;
    // window max includes the center, so v >= m  <=>  v == max (ties kept)
    if (cen.x > THRV && cen.x >= m0) emit(cen.x, flat0 + 0);
    if (cen.y > THRV && cen.y >= m1) emit(cen.y, flat0 + 1);
    if (cen.z > THRV && cen.z >= m2) emit(cen.z, flat0 + 2);
    if (cen.w > THRV && cen.w >= m3) emit(cen.w, flat0 + 3);
    __syncthreads();
}

// ---------------------------------------------------------------------------
__global__ void k_zero(uint32_t* __restrict__ p, int n) {
    int i = blockIdx.x * blockDim.x + threadIdx.x;
    if (i < n) p[i] = 0u;
}

// Pass 1: per-batch histogram of score bits (top 16 bits of positive float).
__global__ void k_hist(const float* __restrict__ hm, uint32_t* __restrict__ hist) {
    __shared__ float4 sig4[NROWS * ROWCH];
    int bi    = blockIdx.x;
    int strip = bi & (NSTRIP - 1);
    int c     = (bi >> 4) % NCLS;
    int b     = bi / (NSTRIP * NCLS);
    uint32_t* h = hist + (size_t)b * NBIN;
    nms_tile(hm, b, c, strip, sig4, [&](float s, int flat) {
        (void)flat;
        uint32_t bin = __float_as_uint(s) >> 16;   // monotone for positive floats
        atomicAdd(&h[bin], 1u);
    });
}

// Pass 2: find cutoff bin (bin of the K-th largest candidate) per batch.
__global__ void k_cutoff(const uint32_t* __restrict__ hist, uint32_t* __restrict__ cutoff) {
    __shared__ uint32_t psum[256];
    const int b = blockIdx.x;
    const int t = threadIdx.x;
    const uint32_t* h = hist + (size_t)b * NBIN;
    const int SEG = NBIN / 256;   // 128 bins per thread
    uint32_t s = 0;
    for (int i = 0; i < SEG; ++i) s += h[t * SEG + i];
    psum[t] = s;
    __syncthreads();
    if (t == 0) {
        uint32_t running = 0;
        int seg = -1;
        for (int q = 255; q >= 0; --q) {
            if (running + psum[q] >= K_TOP) { seg = q; break; }
            running += psum[q];
        }
        uint32_t cb = 0;   // total < K -> collect everything (count < K <= CAP)
        if (seg >= 0) {
            for (int bin = seg * SEG + SEG - 1; bin >= seg * SEG; --bin) {
                running += h[bin];
                if (running >= K_TOP) { cb = (uint32_t)bin; break; }
            }
        }
        cutoff[b] = cb;
    }
}

// Pass 3: collect candidates with bin >= cutoff as 64-bit sort keys.
__global__ void k_collect(const float* __restrict__ hm,
                          const uint32_t* __restrict__ cutoff,
                          uint32_t* __restrict__ cnt,
                          uint64_t* __restrict__ keys) {
    __shared__ float4 sig4[NROWS * ROWCH];
    int bi    = blockIdx.x;
    int strip = bi & (NSTRIP - 1);
    int c     = (bi >> 4) % NCLS;
    int b     = bi / (NSTRIP * NCLS);
    const uint32_t cb = cutoff[b];
    nms_tile(hm, b, c, strip, sig4, [&](float s, int flat) {
        uint32_t bits = __float_as_uint(s);
        if ((bits >> 16) >= cb) {
            uint32_t pos = atomicAdd(&cnt[b], 1u);
            if (pos < CAP) {
                // key: higher score wins; on bit-equal scores, lower flat index wins
                keys[(size_t)b * CAP + pos] =
                    ((uint64_t)bits << 32) | (uint32_t)(~(uint32_t)flat);
            }
        }
    });
}

// Pass 4: bitonic sort (descending) + decode + write outputs.
__global__ void __launch_bounds__(1024)
k_final(const uint32_t* __restrict__ cnt, const uint64_t* __restrict__ keys,
        const float* __restrict__ off, const float* __restrict__ wh,
        float* __restrict__ out) {
    __shared__ unsigned long long a[SORTN];   // 32 KB
    const int b = blockIdx.x;
    const int t = threadIdx.x;
    uint32_t n = cnt[b];
    if (n > CAP) n = CAP;
    for (int i = t; i < SORTN; i += 1024)
        a[i] = (i < (int)n) ? (unsigned long long)keys[(size_t)b * CAP + i] : 0ull;
    __syncthreads();

    for (int k = 2; k <= SORTN; k <<= 1) {
        for (int j = k >> 1; j > 0; j >>= 1) {
            for (int i = t; i < SORTN; i += 1024) {
                int p = i ^ j;
                if (p > i) {
                    bool up = ((i & k) == 0);          // descending overall
                    unsigned long long x = a[i], y = a[p];
                    if ((x < y) == up) { a[i] = y; a[p] = x; }
                }
            }
            __syncthreads();
        }
    }

    if (t < K_TOP) {
        unsigned long long key = a[t];
        float score = -1.0f, cls = -1.0f;
        float x1 = -1.0f, y1 = -1.0f, x2 = -1.0f, y2 = -1.0f;
        if (key != 0ull) {
            uint32_t bits = (uint32_t)(key >> 32);
            uint32_t flat = ~(uint32_t)(key & 0xFFFFFFFFu);
            float s = __uint_as_float(bits);
            uint32_t c   = flat / HW;
            uint32_t idx = flat % HW;
            float ys = (float)(idx / WW);
            float xs = (float)(idx % WW);
            const float* ob = off + (size_t)b * 2 * HW;
            const float* wb = wh  + (size_t)b * 2 * HW;
            float ox = ob[idx], oy = ob[HW + idx];
            float sw = wb[idx], sh = wb[HW + idx];
            xs += ox; ys += oy;
            x1 = fmaxf((xs - sw * 0.5f) * STRIDEF, 0.0f);
            y1 = fmaxf((ys - sh * 0.5f) * STRIDEF, 0.0f);
            x2 = fminf((xs + sw * 0.5f) * STRIDEF, IMGW - 1.0f);
            y2 = fminf((ys + sh * 0.5f) * STRIDEF, IMGH - 1.0f);
            score = s; cls = (float)c;
        }
        const int o = b * K_TOP + t;
        out[o] = score;                                   // scores  [16,100]
        out[BATCH * K_TOP + o] = cls;                     // classes [16,100]
        float* bb = out + 2 * BATCH * K_TOP + (size_t)o * 4;  // bboxes [16,100,4]
        bb[0] = x1; bb[1] = y1; bb[2] = x2; bb[3] = y2;
    }
}

// ---------------------------------------------------------------------------
extern "C" void kernel_launch(void* const* d_in, const int* in_sizes, int n_in,
                              void* d_out, int out_size, void* d_ws, size_t ws_size,
                              hipStream_t stream) {
    (void)in_sizes; (void)n_in; (void)out_size; (void)ws_size;
    const float* hm  = (const float*)d_in[0];
    const float* off = (const float*)d_in[1];
    const float* wh  = (const float*)d_in[2];
    float* out = (float*)d_out;

    // workspace layout
    uint32_t* hist   = (uint32_t*)d_ws;                  // BATCH*NBIN
    uint32_t* cutoff = hist + (size_t)BATCH * NBIN;      // BATCH
    uint32_t* cnt    = cutoff + BATCH;                   // BATCH
    uint64_t* keys   = (uint64_t*)(cnt + BATCH);         // BATCH*CAP (8B aligned)

    const int nzero = BATCH * NBIN + 2 * BATCH;          // hist + cutoff + cnt
    k_zero<<<(nzero + 255) / 256, 256, 0, stream>>>(hist, nzero);

    const int nblk = BATCH * NCLS * NSTRIP;              // 20480
    k_hist  <<<nblk,  256, 0, stream>>>(hm, hist);
    k_cutoff<<<BATCH, 256, 0, stream>>>(hist, cutoff);
    k_collect<<<nblk, 256, 0, stream>>>(hm, cutoff, cnt, keys);
    k_final <<<BATCH, 1024, 0, stream>>>(cnt, keys, off, wh, out);
}